// ResiduesDecodeVec_28028956574042
// MI455X (gfx1250) — compile-verified
//
#include <hip/hip_runtime.h>

// CDNA5 / gfx1250: wave32, v_wmma_f32_16x16x4_f32 (full f32 precision).
// Each block = (token s, 16-batch slab) x all 3 vector components:
//  - x tile load is one contiguous 48KB stream (coalesced)
//  - each Wh/Wo B-fragment feeds 3 WMMAs (d=0,1,2) -> 3x less L2 traffic,
//    3 independent accumulator chains for back-to-back WMMA issue.

typedef __attribute__((ext_vector_type(2))) float v2f;
typedef __attribute__((ext_vector_type(8))) float v8f;

__constant__ int RES_LEN_D[20] = {4,10,7,7,5,8,8,3,9,7,7,8,7,10,6,5,6,13,11,6};

#define SEQ    512
#define VDIM   256
#define XPITCH 260   // (256+4) f32: 8B-aligned rows, conflict-free ds_load_b64

// base[s] = first ragged-output index of token s (all tokens have len >= 3)
__global__ void token_base_kernel(const int* __restrict__ seq_idx,
                                  int* __restrict__ base, int TA) {
  int t = blockIdx.x * blockDim.x + threadIdx.x;
  if (t >= TA) return;
  int s = seq_idx[t];
  if (t == 0 || seq_idx[t - 1] != s) base[s] = t;
}

__launch_bounds__(256, 2)
__global__ void moe_vec_decode_kernel(const float* __restrict__ x,    // (64,512,256,3)
                                      const float* __restrict__ Wh,   // (20,256,256)
                                      const float* __restrict__ Wo,   // (20,13,256)
                                      const int*   __restrict__ res_ids,
                                      const int*   __restrict__ base,
                                      float*       __restrict__ out,  // (64,TA,3)
                                      int TA) {
  // M = 48 rows: m = d*16 + i  (i = batch-in-slab, d = vector component)
  __shared__ float Xs[48][XPITCH];     // X tile (48 x 256)
  __shared__ float Hs[48][XPITCH];     // H tile (48 x 256)
  // stage-2 partials alias Xs (dead after stage 1; barrier-separated)
  float (*Ps)[3][16][16] = (float (*)[3][16][16])&Xs[0][0];  // [8][3][16][16]

  const int s   = blockIdx.y;          // token
  const int b0  = blockIdx.x << 4;     // batch slab base: 0,16,32,48
  const int r   = res_ids[s];
  const int tb  = base[s];
  const int len = RES_LEN_D[r];

  const int tid  = threadIdx.x;
  const int lane = tid & 31;
  const int w    = tid >> 5;           // wave 0..7
  const int half = lane >> 4;          // K half-select (ISA f32 A/B layout)
  const int l16  = lane & 15;

  // ---- load x[b0..b0+15, s, :, :] (16 x 768 contiguous f32) into LDS ----
  {
    const float* xb = x + ((size_t)b0 * SEQ + s) * VDIM * 3;
    // 12288 floats, 256 threads x 48 fully-coalesced b32 loads
    #pragma unroll
    for (int it = 0; it < 48; ++it) {
      const int g = it * 256 + tid;        // 0..12287
      const int i = g / 768;               // batch row in slab
      const int j = g - i * 768;           // 0..767
      const int k = j / 3;                 // channel
      const int d = j - k * 3;             // vector component
      Xs[d * 16 + i][k] = xb[(size_t)i * 768 + j];
    }
  }
  __syncthreads();

  // ---- stage 1: H = X @ Wh[r]^T ; B[k][n] = Wh[r][n][k] ----
  const float* WhR = Wh + (size_t)r * VDIM * VDIM;
  #pragma unroll
  for (int jj = 0; jj < 2; ++jj) {
    const int N0 = (w << 5) + (jj << 4);
    const float* brow = WhR + (size_t)(N0 + l16) * VDIM;   // lane's B column
    v8f a0 = {}, a1 = {}, a2 = {};
    #pragma unroll 8
    for (int kk = 0; kk < 64; ++kk) {                      // K = 256 = 64 x 4
      const int k0 = (kk << 2) + (half << 1);
      v2f bf = *(const v2f*)(brow + k0);                   // shared B fragment
      v2f f0 = *(const v2f*)&Xs[l16     ][k0];             // d = 0
      v2f f1 = *(const v2f*)&Xs[l16 + 16][k0];             // d = 1
      v2f f2 = *(const v2f*)&Xs[l16 + 32][k0];             // d = 2
      a0 = __builtin_amdgcn_wmma_f32_16x16x4_f32(false, f0, false, bf, (short)0, a0, false, false);
      a1 = __builtin_amdgcn_wmma_f32_16x16x4_f32(false, f1, false, bf, (short)0, a1, false, false);
      a2 = __builtin_amdgcn_wmma_f32_16x16x4_f32(false, f2, false, bf, (short)0, a2, false, false);
    }
    #pragma unroll
    for (int j = 0; j < 8; ++j) {                          // C: VGPR j -> M=j, j+8
      const int mr = j + (half << 3);
      Hs[mr     ][N0 + l16] = a0[j];
      Hs[mr + 16][N0 + l16] = a1[j];
      Hs[mr + 32][N0 + l16] = a2[j];
    }
  }
  __syncthreads();

  // ---- stage 2: O = H @ Wo[r]^T, K=256 split across 8 waves (32 each) ----
  {
    const float* WoR  = Wo + (size_t)r * 13 * VDIM;
    const float* brow = WoR + (size_t)(l16 < 13 ? l16 : 12) * VDIM;
    const int K0w = w << 5;
    v8f a0 = {}, a1 = {}, a2 = {};
    #pragma unroll
    for (int kk = 0; kk < 8; ++kk) {
      const int k0 = K0w + (kk << 2) + (half << 1);
      v2f bf = *(const v2f*)(brow + k0);
      if (l16 >= 13) { bf[0] = 0.f; bf[1] = 0.f; }         // pad columns -> 0
      v2f f0 = *(const v2f*)&Hs[l16     ][k0];
      v2f f1 = *(const v2f*)&Hs[l16 + 16][k0];
      v2f f2 = *(const v2f*)&Hs[l16 + 32][k0];
      a0 = __builtin_amdgcn_wmma_f32_16x16x4_f32(false, f0, false, bf, (short)0, a0, false, false);
      a1 = __builtin_amdgcn_wmma_f32_16x16x4_f32(false, f1, false, bf, (short)0, a1, false, false);
      a2 = __builtin_amdgcn_wmma_f32_16x16x4_f32(false, f2, false, bf, (short)0, a2, false, false);
    }
    #pragma unroll
    for (int j = 0; j < 8; ++j) {
      const int mr = j + (half << 3);
      Ps[w][0][mr][l16] = a0[j];
      Ps[w][1][mr][l16] = a1[j];
      Ps[w][2][mr][l16] = a2[j];
    }
  }
  __syncthreads();

  // ---- reduce 8 partials per (d,i,a), scatter valid atoms ----
  {
    const int i = tid >> 4;     // batch row in slab
    const int a = tid & 15;     // atom index
    #pragma unroll
    for (int d = 0; d < 3; ++d) {
      float sum = 0.f;
      #pragma unroll
      for (int ww = 0; ww < 8; ++ww) sum += Ps[ww][d][i][a];
      if (a < len)
        out[((size_t)(b0 + i) * TA + tb + a) * 3 + d] = sum;
    }
  }
}

extern "C" void kernel_launch(void* const* d_in, const int* in_sizes, int n_in,
                              void* d_out, int out_size, void* d_ws, size_t ws_size,
                              hipStream_t stream) {
  const float* x       = (const float*)d_in[0];
  const float* Wh      = (const float*)d_in[1];
  const float* Wo      = (const float*)d_in[2];
  const int*   res_ids = (const int*)d_in[3];
  const int*   seq_idx = (const int*)d_in[4];
  // d_in[5] (atom_idx) not needed: atoms within a token are contiguous 0..len-1
  const int TA = in_sizes[4];          // total ragged atoms

  int* base = (int*)d_ws;              // 512 ints of scratch, rewritten every call
  token_base_kernel<<<(TA + 255) / 256, 256, 0, stream>>>(seq_idx, base, TA);

  dim3 grid(4, SEQ);                   // 4 batch slabs x 512 tokens
  moe_vec_decode_kernel<<<grid, 256, 0, stream>>>(x, Wh, Wo, res_ids, base,
                                                  (float*)d_out, TA);
}